// GCN_raw_att_12_68968584839878
// MI455X (gfx1250) — compile-verified
//
#include <hip/hip_runtime.h>
#include <hip/hip_bf16.h>

#define NN 50000
#define NE 800000
#define NG 64

typedef __attribute__((ext_vector_type(2))) float v2f;
typedef __attribute__((ext_vector_type(8))) float v8f;

// ---------------------------------------------------------------- degree
__global__ void k_count(const int* __restrict__ dst, float* __restrict__ cnt, int E) {
    int gid = blockIdx.x * blockDim.x + threadIdx.x;
    if (gid < E) atomicAdd(&cnt[dst[gid]], 1.0f);
}
__global__ void k_invert(float* __restrict__ cnt, int N) {
    int gid = blockIdx.x * blockDim.x + threadIdx.x;
    if (gid < N) cnt[gid] = 1.0f / fmaxf(cnt[gid], 1.0f);
}

// ---------------------------------------------------------------- edge scatter: agg[dst] += x[src]*ew
template <int DIN>
__global__ void k_scatter(const float* __restrict__ h, const int* __restrict__ src,
                          const int* __restrict__ dst, const float* __restrict__ ew,
                          float* __restrict__ agg) {
    constexpr int CHUNKS = (DIN + 3) / 4;                 // 1,8,16,32,64 -> all pow2
    constexpr int LOG2C = (CHUNKS == 1) ? 0 : (CHUNKS == 8) ? 3
                        : (CHUNKS == 16) ? 4 : (CHUNKS == 32) ? 5 : 6;
    long long gid = (long long)blockIdx.x * blockDim.x + threadIdx.x;
    if (gid >= (long long)NE * CHUNKS) return;
    int e = (int)(gid >> LOG2C);
    int c = (int)(gid & (CHUNKS - 1));
    int s = src[e], d = dst[e];
    float w = ew[e];
    if constexpr (DIN % 4 == 0) {
        const float4 v = *(const float4*)(h + (size_t)s * DIN + c * 4);
        float* ap = agg + (size_t)d * DIN + c * 4;
        atomicAdd(ap + 0, v.x * w);
        atomicAdd(ap + 1, v.y * w);
        atomicAdd(ap + 2, v.z * w);
        atomicAdd(ap + 3, v.w * w);
    } else {                                              // DIN==3 first layer
        const float* hp = h + (size_t)s * DIN;
        float* ap = agg + (size_t)d * DIN;
#pragma unroll
        for (int u = 0; u < DIN; ++u) atomicAdd(ap + u, hp[u] * w);
    }
}

// ---------------------------------------------------------------- fused GraphConv GEMM + BN + ReLU (fp32 WMMA)
// hout[n,j] = relu( bn( (agg[n,:]*deg[n]) @ Wrel + brel + hin[n,:] @ Wroot ) )
template <int DIN, int DOUT>
__launch_bounds__(128)
__global__ void k_conv_wmma(const float* __restrict__ agg, const float* __restrict__ deg,
                            const float* __restrict__ hin,
                            const float* __restrict__ Wrel, const float* __restrict__ brel,
                            const float* __restrict__ Wroot,
                            const float* __restrict__ gamma, const float* __restrict__ beta,
                            const float* __restrict__ mean, const float* __restrict__ var,
                            float* __restrict__ hout) {
    constexpr int DINP = (DIN < 4) ? 4 : DIN;             // pad K for din==3
    __shared__ __align__(16) float lA[16 * DINP];         // scaled aggregate tile
    __shared__ __align__(16) float lX[16 * DINP];         // root (x) tile
    const int tid = threadIdx.x;
    const int wave = tid >> 5;
    const int lane = tid & 31;
    const int rowBase = blockIdx.x * 16;

    if constexpr (DIN % 4 == 0) {                         // vectorized staging (b128)
        constexpr int C4 = DINP / 4;
        for (int i = tid; i < 16 * C4; i += 128) {
            int r = i / C4, c4 = i % C4;                  // pow2 -> shifts
            int row = rowBase + r;
            float dg = deg[row];
            const float4 a4 = *(const float4*)(agg + (size_t)row * DIN + c4 * 4);
            const float4 x4 = *(const float4*)(hin + (size_t)row * DIN + c4 * 4);
            *(float4*)(lA + r * DINP + c4 * 4) = make_float4(a4.x * dg, a4.y * dg, a4.z * dg, a4.w * dg);
            *(float4*)(lX + r * DINP + c4 * 4) = x4;
        }
    } else {                                              // din==3: scalar, zero-pad K=3
        for (int i = tid; i < 16 * DINP; i += 128) {
            int r = i / DINP, cc = i % DINP;
            int row = rowBase + r;
            float av = 0.0f, xv = 0.0f;
            if (cc < DIN) {
                av = agg[(size_t)row * DIN + cc] * deg[row];
                xv = hin[(size_t)row * DIN + cc];
            }
            lA[r * DINP + cc] = av;
            lX[r * DINP + cc] = xv;
        }
    }
    __syncthreads();

    const int colBase = (blockIdx.y * 4 + wave) * 16;     // one 16x16 tile per wave
    if (colBase < DOUT) {                                 // wave-uniform; EXEC all-1s for WMMA
        const int half = lane >> 4;                       // lanes16-31 hold K=k+2,k+3
        const int mr = lane & 15;
        const int col = colBase + mr;
        __builtin_prefetch(Wrel + col, 0, 2);             // global_prefetch_b8 on the B panels
        __builtin_prefetch(Wroot + col, 0, 2);
        v8f c = {};
#pragma unroll 8
        for (int k = 0; k < DINP; k += 4) {
            const int kk = k + 2 * half;
            int k0, k1;
            if constexpr (DIN % 4 == 0) { k0 = kk; k1 = kk + 1; }
            else { k0 = kk < DIN ? kk : DIN - 1; k1 = (kk + 1) < DIN ? kk + 1 : DIN - 1; }
            v2f a, x, br, bt;
            a.x = lA[mr * DINP + kk];  a.y = lA[mr * DINP + kk + 1];
            x.x = lX[mr * DINP + kk];  x.y = lX[mr * DINP + kk + 1];
            br.x = Wrel[(size_t)k0 * DOUT + col];
            br.y = Wrel[(size_t)k1 * DOUT + col];
            bt.x = Wroot[(size_t)k0 * DOUT + col];
            bt.y = Wroot[(size_t)k1 * DOUT + col];
            c = __builtin_amdgcn_wmma_f32_16x16x4_f32(false, a, false, br, (short)0, c, false, false);
            c = __builtin_amdgcn_wmma_f32_16x16x4_f32(false, x, false, bt, (short)0, c, false, false);
        }
        // fused b_rel + BatchNorm(eval) + ReLU epilogue
        float sc = gamma[col] * rsqrtf(var[col] + 1e-5f);
        float sh = beta[col] - mean[col] * sc;
        float bb = brel[col];
#pragma unroll
        for (int r = 0; r < 8; ++r) {                     // C: VGPR r -> row r+8*half, col=lane&15
            int row = rowBase + r + half * 8;
            hout[(size_t)row * DOUT + col] = fmaxf((c[r] + bb) * sc + sh, 0.0f);
        }
    }
}

// ---------------------------------------------------------------- graph mean pool (F = 512)
__global__ void k_pool_sum(const float* __restrict__ h, const int* __restrict__ batch,
                           float* __restrict__ gsum) {
    long long gid = (long long)blockIdx.x * blockDim.x + threadIdx.x;
    if (gid >= (long long)NN * 512) return;
    int n = (int)(gid >> 9);
    int f = (int)(gid & 511);
    atomicAdd(&gsum[((size_t)batch[n] << 9) + f], h[gid]);
}
__global__ void k_gcnt(const int* __restrict__ batch, float* __restrict__ gcnt, int N) {
    int gid = blockIdx.x * blockDim.x + threadIdx.x;
    if (gid < N) atomicAdd(&gcnt[batch[gid]], 1.0f);
}
__global__ void k_gmean(const float* __restrict__ gsum, const float* __restrict__ gcnt,
                        float* __restrict__ g) {
    int gid = blockIdx.x * blockDim.x + threadIdx.x;
    if (gid < NG * 512) g[gid] = gsum[gid] / fmaxf(gcnt[gid >> 9], 1.0f);
}

// ---------------------------------------------------------------- tiny FC layers (64 rows; VALU)
__global__ void k_fc_bn_relu(const float* __restrict__ gin, const float* __restrict__ W,
                             const float* __restrict__ b,
                             const float* __restrict__ gamma, const float* __restrict__ beta,
                             const float* __restrict__ mean, const float* __restrict__ var,
                             float* __restrict__ gout, int G, int din, int dout) {
    int gid = blockIdx.x * blockDim.x + threadIdx.x;
    if (gid >= G * dout) return;
    int g = gid / dout, j = gid % dout;
    float acc = b[j];
    for (int k = 0; k < din; ++k) acc += gin[g * din + k] * W[k * dout + j];
    float sc = gamma[j] * rsqrtf(var[j] + 1e-5f);
    gout[gid] = fmaxf(0.0f, (acc - mean[j]) * sc + beta[j]);
}
__global__ void k_fc4_lsm(const float* __restrict__ gin, const float* __restrict__ W,
                          const float* __restrict__ b, float* __restrict__ out, int G, int din) {
    int g = blockIdx.x * blockDim.x + threadIdx.x;
    if (g >= G) return;
    float l0 = b[0], l1 = b[1];
    for (int k = 0; k < din; ++k) {
        float v = gin[g * din + k];
        l0 += v * W[k * 2 + 0];
        l1 += v * W[k * 2 + 1];
    }
    float m = fmaxf(l0, l1);
    float lse = m + logf(expf(l0 - m) + expf(l1 - m));
    out[g * 2 + 0] = l0 - lse;
    out[g * 2 + 1] = l1 - lse;
}

// ---------------------------------------------------------------- per-layer runner
template <int DIN, int DOUT>
static inline void run_conv(const float* hin, float* hout, float* agg, const float* deg,
                            const int* src, const int* dst, const float* ew,
                            void* const* d_in, int base, hipStream_t stream) {
    hipMemsetAsync(agg, 0, (size_t)NN * DIN * sizeof(float), stream);
    constexpr long long total = (long long)NE * ((DIN + 3) / 4);
    k_scatter<DIN><<<(unsigned)((total + 255) / 256), 256, 0, stream>>>(hin, src, dst, ew, agg);
    dim3 grid(NN / 16, (DOUT + 63) / 64);
    k_conv_wmma<DIN, DOUT><<<grid, 128, 0, stream>>>(agg, deg, hin,
        (const float*)d_in[base + 0], (const float*)d_in[base + 1], (const float*)d_in[base + 2],
        (const float*)d_in[base + 3], (const float*)d_in[base + 4],
        (const float*)d_in[base + 5], (const float*)d_in[base + 6], hout);
}

// ---------------------------------------------------------------- launch
extern "C" void kernel_launch(void* const* d_in, const int* in_sizes, int n_in,
                              void* d_out, int out_size, void* d_ws, size_t ws_size,
                              hipStream_t stream) {
    const float* x    = (const float*)d_in[0];
    const int*  eidx  = (const int*)d_in[1];
    const int*  src   = eidx;
    const int*  dst   = eidx + NE;
    const float* ew   = (const float*)d_in[2];
    const int*  batch = (const int*)d_in[3];
    // params (dict order): conv_i{W_rel,b_rel,W_root}+bn_i{g,b,m,v} -> 7 each from idx 4
    //                      fc_j{W,b}+bn_fc_j{g,b,m,v} -> 6 each from idx 39; fc4 at 57
    float* ws   = (float*)d_ws;
    float* agg  = ws;                               // NN*512
    float* hA   = agg  + (size_t)NN * 512;          // NN*512
    float* hB   = hA   + (size_t)NN * 512;          // NN*512
    float* deg  = hB   + (size_t)NN * 512;          // NN
    float* gsum = deg  + NN;                        // NG*512
    float* gcnt = gsum + (size_t)NG * 512;          // NG
    float* gb1  = gcnt + NG;                        // NG*512
    float* gb2  = gb1  + (size_t)NG * 512;          // NG*512

    // in-degrees -> 1/max(cnt,1)
    hipMemsetAsync(deg, 0, NN * sizeof(float), stream);
    k_count<<<(NE + 255) / 256, 256, 0, stream>>>(dst, deg, NE);
    k_invert<<<(NN + 255) / 256, 256, 0, stream>>>(deg, NN);

    run_conv<3,   32 >(x,  hA, agg, deg, src, dst, ew, d_in, 4 + 0 * 7, stream);
    run_conv<32,  64 >(hA, hB, agg, deg, src, dst, ew, d_in, 4 + 1 * 7, stream);
    run_conv<64,  128>(hB, hA, agg, deg, src, dst, ew, d_in, 4 + 2 * 7, stream);
    run_conv<128, 256>(hA, hB, agg, deg, src, dst, ew, d_in, 4 + 3 * 7, stream);
    run_conv<256, 512>(hB, hA, agg, deg, src, dst, ew, d_in, 4 + 4 * 7, stream);

    // graph mean pool [NG,512] from hA
    hipMemsetAsync(gsum, 0, (size_t)NG * 512 * sizeof(float), stream);
    hipMemsetAsync(gcnt, 0, NG * sizeof(float), stream);
    k_pool_sum<<<(unsigned)(((long long)NN * 512 + 255) / 256), 256, 0, stream>>>(hA, batch, gsum);
    k_gcnt<<<(NN + 255) / 256, 256, 0, stream>>>(batch, gcnt, NN);
    k_gmean<<<(NG * 512 + 255) / 256, 256, 0, stream>>>(gsum, gcnt, gb1);

    // FC stack
    const int fdims[3][2] = {{512, 256}, {256, 128}, {128, 32}};
    float* fin = gb1;
    float* fout = gb2;
    for (int j = 0; j < 3; ++j) {
        int base = 39 + j * 6;
        k_fc_bn_relu<<<(NG * fdims[j][1] + 255) / 256, 256, 0, stream>>>(fin,
            (const float*)d_in[base + 0], (const float*)d_in[base + 1],
            (const float*)d_in[base + 2], (const float*)d_in[base + 3],
            (const float*)d_in[base + 4], (const float*)d_in[base + 5],
            fout, NG, fdims[j][0], fdims[j][1]);
        float* t = fin; fin = fout; fout = t;
    }
    k_fc4_lsm<<<1, NG, 0, stream>>>(fin, (const float*)d_in[57], (const float*)d_in[58],
                                    (float*)d_out, NG, 32);
}